// JLModel_39367670235338
// MI455X (gfx1250) — compile-verified
//
#include <hip/hip_runtime.h>
#include <math.h>

// -------- problem sizes (match reference) ----------------------------------
#define BATCH 8192
#define D_IN  1024
#define D_OUT 1024
#define NP    2048

typedef __attribute__((ext_vector_type(2))) float v2f;
typedef __attribute__((ext_vector_type(8))) float v8f;
typedef __attribute__((ext_vector_type(4))) int   v4i;

// -------- gfx1250 async global->LDS path (guarded) -------------------------
// Diagnostic-confirmed signatures:
//   __builtin_amdgcn_global_load_async_to_lds_b128(AS1 v4i* src, AS3 dst, int off, int cpol)
//   __builtin_amdgcn_global_load_async_to_lds_b32 (AS1 int* src, AS3 dst, int off, int cpol)
#if __has_builtin(__builtin_amdgcn_global_load_async_to_lds_b128) && \
    __has_builtin(__builtin_amdgcn_global_load_async_to_lds_b32)  && \
    __has_builtin(__builtin_amdgcn_s_wait_asynccnt)
#define HAVE_ASYNC_LDS 1
typedef __attribute__((address_space(1))) int g_i32;
typedef __attribute__((address_space(1))) v4i g_v4i;
typedef __attribute__((address_space(3))) int l_i32;
typedef __attribute__((address_space(3))) v4i l_v4i;
#define ASYNC_LDS_B128(src, dst) \
  __builtin_amdgcn_global_load_async_to_lds_b128((g_v4i*)(src), (l_v4i*)(dst), 0, 0)
#define ASYNC_LDS_B32(src, dst) \
  __builtin_amdgcn_global_load_async_to_lds_b32((g_i32*)(src), (l_i32*)(dst), 0, 0)
#endif

#define TM 128
#define TN 128
#define TK 32

// ===========================================================================
// Kernel 1: Out = X @ W + b   (8192x1024 * 1024x1024, fp32 WMMA 16x16x4)
// 256 threads = 8 waves -> 128x128 tile; wave w owns a 16x128 strip.
// LDS tiles stored K-innermost so every WMMA fragment is one aligned v2f
// (single ds_load_b64 into an even register pair, no repacking movs).
// Double-buffered; staged by async global->LDS DMA when available.
// ===========================================================================
__global__ __launch_bounds__(256) void gemm_xw_wmma(const float* __restrict__ X,
                                                    const float* __restrict__ Wm,
                                                    const float* __restrict__ bias,
                                                    float* __restrict__ Out) {
  __shared__ float As[2][TM][TK];   // As[m][k] = X[m0+m][k0+k]   (row-major copy)
  __shared__ float Bs[2][TN][TK];   // Bs[n][k] = W[k0+k][n0+n]   (transposed copy)
  const int tid  = threadIdx.x;
  const int wave = tid >> 5;
  const int lane = tid & 31;
  const int l16  = lane & 15;
  const int kh   = (lane >> 4) << 1;   // lanes 0-15 -> K {0,1}; lanes 16-31 -> K {2,3}
  const int m0   = blockIdx.y * TM;
  const int n0   = blockIdx.x * TN;

  v8f acc[8] = {};

  auto stage = [&](int buf, int k0) {
#if HAVE_ASYNC_LDS
    for (int i = tid * 4; i < TM * TK; i += 256 * 4) {       // A: 16B chunks along k
      const int m = i >> 5, k = i & (TK - 1);
      ASYNC_LDS_B128(X + (size_t)(m0 + m) * D_IN + (k0 + k), &As[buf][m][k]);
    }
    for (int i = tid; i < TK * TN; i += 256) {               // B: DMA-side transpose
      const int k = i >> 7, n = i & (TN - 1);
      ASYNC_LDS_B32(Wm + (size_t)(k0 + k) * D_OUT + (n0 + n), &Bs[buf][n][k]);
    }
#else
    for (int i = tid; i < TM * TK; i += 256) {
      const int m = i >> 5, k = i & (TK - 1);
      As[buf][m][k] = X[(size_t)(m0 + m) * D_IN + (k0 + k)];
    }
    for (int i = tid; i < TK * TN; i += 256) {
      const int k = i >> 7, n = i & (TN - 1);
      Bs[buf][n][k] = Wm[(size_t)(k0 + k) * D_OUT + (n0 + n)];
    }
#endif
  };
  auto stage_wait = [&]() {
#if HAVE_ASYNC_LDS
    __builtin_amdgcn_s_wait_asynccnt(0);
#endif
    __syncthreads();
  };

  const int NT = D_IN / TK;
  stage(0, 0);
  stage_wait();
  for (int kt = 0; kt < NT; ++kt) {
    const int cur = kt & 1;
    if (kt + 1 < NT) stage(cur ^ 1, (kt + 1) * TK);          // DMA next, overlap compute
#pragma unroll
    for (int kk = 0; kk < TK; kk += 4) {
      const v2f a = *(const v2f*)&As[cur][wave * 16 + l16][kk + kh];
#pragma unroll
      for (int t = 0; t < 8; ++t) {
        const v2f b = *(const v2f*)&Bs[cur][t * 16 + l16][kk + kh];
        acc[t] = __builtin_amdgcn_wmma_f32_16x16x4_f32(
            false, a, false, b, (short)0, acc[t], false, false);
      }
    }
    stage_wait();
  }
  // C layout: VGPR r -> (M = r + 8*(lane>=16), N = lane&15)
  const int mo = (lane >> 4) << 3;
#pragma unroll
  for (int t = 0; t < 8; ++t) {
    const int gn = n0 + t * 16 + l16;
    const float bv = bias[gn];
#pragma unroll
    for (int r = 0; r < 8; ++r) {
      const int gm = m0 + wave * 16 + mo + r;
      Out[(size_t)gm * D_OUT + gn] = acc[t][r] + bv;
    }
  }
}

// ===========================================================================
// Kernel 2: F = (G^T G) / BATCH   (G: 8192x2048, F: 2048x2048)
// Both operand tiles come from G and both need column access -> both stored
// transposed (K-innermost) via per-element async DMA.
// ===========================================================================
__global__ __launch_bounds__(256) void gemm_fisher_wmma(const float* __restrict__ G,
                                                        float* __restrict__ F) {
  __shared__ float GiT[2][TM][TK];  // GiT[m][k] = G[k0+k][i0+m]
  __shared__ float GjT[2][TN][TK];  // GjT[n][k] = G[k0+k][j0+n]
  const int tid  = threadIdx.x;
  const int wave = tid >> 5;
  const int lane = tid & 31;
  const int l16  = lane & 15;
  const int kh   = (lane >> 4) << 1;
  const int i0   = blockIdx.y * TM;
  const int j0   = blockIdx.x * TN;

  v8f acc[8] = {};

  auto stage = [&](int buf, int k0) {
#if HAVE_ASYNC_LDS
    for (int i = tid; i < TK * TM; i += 256) {
      const int k = i >> 7, c = i & (TM - 1);
      ASYNC_LDS_B32(G + (size_t)(k0 + k) * NP + (i0 + c), &GiT[buf][c][k]);
    }
    for (int i = tid; i < TK * TN; i += 256) {
      const int k = i >> 7, c = i & (TN - 1);
      ASYNC_LDS_B32(G + (size_t)(k0 + k) * NP + (j0 + c), &GjT[buf][c][k]);
    }
#else
    for (int i = tid; i < TK * TM; i += 256) {
      const int k = i >> 7, c = i & (TM - 1);
      GiT[buf][c][k] = G[(size_t)(k0 + k) * NP + (i0 + c)];
    }
    for (int i = tid; i < TK * TN; i += 256) {
      const int k = i >> 7, c = i & (TN - 1);
      GjT[buf][c][k] = G[(size_t)(k0 + k) * NP + (j0 + c)];
    }
#endif
  };
  auto stage_wait = [&]() {
#if HAVE_ASYNC_LDS
    __builtin_amdgcn_s_wait_asynccnt(0);
#endif
    __syncthreads();
  };

  const int NT = BATCH / TK;
  stage(0, 0);
  stage_wait();
  for (int kt = 0; kt < NT; ++kt) {
    const int cur = kt & 1;
    if (kt + 1 < NT) stage(cur ^ 1, (kt + 1) * TK);
#pragma unroll
    for (int kk = 0; kk < TK; kk += 4) {
      const v2f a = *(const v2f*)&GiT[cur][wave * 16 + l16][kk + kh];
#pragma unroll
      for (int t = 0; t < 8; ++t) {
        const v2f b = *(const v2f*)&GjT[cur][t * 16 + l16][kk + kh];
        acc[t] = __builtin_amdgcn_wmma_f32_16x16x4_f32(
            false, a, false, b, (short)0, acc[t], false, false);
      }
    }
    stage_wait();
  }
  const float inv_b = 1.0f / (float)BATCH;
  const int mo = (lane >> 4) << 3;
#pragma unroll
  for (int t = 0; t < 8; ++t) {
    const int gj = j0 + t * 16 + l16;
#pragma unroll
    for (int r = 0; r < 8; ++r) {
      const int gi = i0 + wave * 16 + mo + r;
      F[(size_t)gi * NP + gj] = acc[t][r] * inv_b;
    }
  }
}

// ===========================================================================
// Power iteration machinery (deterministic: no atomics).
//   phase 1: lambda_max(F);  phase 2: lambda_max(c*I - F) -> lambda_min(F)
// ===========================================================================

// y[row] = (shifted ? c*v[row] - (F v)[row] : (F v)[row]); one block per row.
__global__ __launch_bounds__(256) void matvec_shift(const float* __restrict__ F,
                                                    const float* __restrict__ v,
                                                    float* __restrict__ y,
                                                    const float* __restrict__ shiftp,
                                                    int shifted) {
  const int row = blockIdx.x;
  const int tid = threadIdx.x;
  const float4* Frow = (const float4*)(F + (size_t)row * NP);
  const float4* v4   = (const float4*)v;
  float dot = 0.0f;
#pragma unroll 2
  for (int j = tid; j < NP / 4; j += 256) {
    float4 f = Frow[j];
    float4 w = v4[j];
    dot += f.x * w.x + f.y * w.y + f.z * w.z + f.w * w.w;
  }
  __shared__ float red[256];
  red[tid] = dot;
  __syncthreads();
  for (int s = 128; s > 0; s >>= 1) {
    if (tid < s) red[tid] += red[tid + s];
    __syncthreads();
  }
  if (tid == 0) {
    float d = red[0];
    y[row] = shifted ? (shiftp[0] * v[row] - d) : d;
  }
}

// Single block: lambda = v.y (Rayleigh, v was unit), then v = y / ||y||.
__global__ __launch_bounds__(256) void reduce_normalize(float* __restrict__ v,
                                                        const float* __restrict__ y,
                                                        float* __restrict__ lambda) {
  const int tid = threadIdx.x;
  __shared__ float s0[256], s1[256];
  __shared__ float sh_scale;
  float p0 = 0.0f, p1 = 0.0f;
  for (int i = tid; i < NP; i += 256) {
    float yi = y[i];
    p0 += v[i] * yi;
    p1 += yi * yi;
  }
  s0[tid] = p0; s1[tid] = p1;
  __syncthreads();
  for (int s = 128; s > 0; s >>= 1) {
    if (tid < s) { s0[tid] += s0[tid + s]; s1[tid] += s1[tid + s]; }
    __syncthreads();
  }
  if (tid == 0) {
    lambda[0] = s0[0];
    sh_scale = 1.0f / (sqrtf(s1[0]) + 1e-30f);
  }
  __syncthreads();
  const float sc = sh_scale;
  for (int i = tid; i < NP; i += 256) v[i] = y[i] * sc;
}

__global__ __launch_bounds__(256) void seed_vec(float* __restrict__ v, unsigned mul,
                                                unsigned add) {
  const int i = blockIdx.x * 256 + threadIdx.x;
  if (i < NP) {
    unsigned h = (unsigned)i * mul + add;
    h ^= h >> 13; h *= 0x5bd1e995u; h ^= h >> 15;
    v[i] = 0.25f + (float)(h & 1023u) * (1.0f / 1024.0f);
  }
}

// shift = 1.08 * lambda_max estimate (safety margin so c*I - F stays PSD-top)
__global__ void prep_phase2(const float* __restrict__ lambda, float* __restrict__ shiftp) {
  shiftp[0] = 1.08f * lambda[0] + 1e-12f;
}

// lambda_min = c - lambda_max(cI - F); reg = 0.1 * max(0.01 - lambda_min, 0)
__global__ void finalize_reg(const float* __restrict__ shiftp,
                             const float* __restrict__ lambdaB,
                             float* __restrict__ out_reg) {
  float lmin = shiftp[0] - lambdaB[0];
  float pen  = fmaxf(0.01f - lmin, 0.0f);
  out_reg[0] = 0.1f * pen;
}

// ===========================================================================
// Host orchestration
// ===========================================================================
extern "C" void kernel_launch(void* const* d_in, const int* in_sizes, int n_in,
                              void* d_out, int out_size, void* d_ws, size_t ws_size,
                              hipStream_t stream) {
  const float* X    = (const float*)d_in[0];   // (8192, 1024)
  const float* G    = (const float*)d_in[1];   // (8192, 2048)
  const float* Wm   = (const float*)d_in[2];   // (1024, 1024)
  const float* bias = (const float*)d_in[3];   // (1024,)
  float* out = (float*)d_out;                  // [8192*1024 output | 1 reg_loss]

  // workspace layout (needs ~16.02 MB)
  char* ws = (char*)d_ws;
  float* F      = (float*)(ws);                                  // 2048*2048*4 = 16 MB
  float* v      = (float*)(ws + (size_t)NP * NP * 4);            // 8 KB
  float* y      = (float*)(ws + (size_t)NP * NP * 4 + NP * 4);   // 8 KB
  float* lam    = (float*)(ws + (size_t)NP * NP * 4 + 2 * NP * 4);
  float* shiftp = lam + 1;

  // 1) forward GEMM: out = X@W + b
  gemm_xw_wmma<<<dim3(D_OUT / TN, BATCH / TM), 256, 0, stream>>>(X, Wm, bias, out);

  // 2) empirical Fisher: F = G^T G / BATCH
  gemm_fisher_wmma<<<dim3(NP / TN, NP / TM), 256, 0, stream>>>(G, F);

  // 3) phase 1: lambda_max(F) via power iteration
  const int T1 = 40, T2 = 80;
  seed_vec<<<NP / 256, 256, 0, stream>>>(v, 1103515245u, 12345u);
  for (int it = 0; it < T1; ++it) {
    matvec_shift<<<NP, 256, 0, stream>>>(F, v, y, shiftp, 0);
    reduce_normalize<<<1, 256, 0, stream>>>(v, y, lam);
  }

  // 4) phase 2: lambda_max(c*I - F), c = 1.08*lambda_max
  prep_phase2<<<1, 1, 0, stream>>>(lam, shiftp);
  seed_vec<<<NP / 256, 256, 0, stream>>>(v, 2654435761u, 40503u);
  for (int it = 0; it < T2; ++it) {
    matvec_shift<<<NP, 256, 0, stream>>>(F, v, y, shiftp, 1);
    reduce_normalize<<<1, 256, 0, stream>>>(v, y, lam);
  }

  // 5) reg_loss = 0.1 * max(0.01 - lambda_min, 0)
  finalize_reg<<<1, 1, 0, stream>>>(shiftp, lam, out + (size_t)BATCH * D_OUT);
}